// OFTTA_89343909691524
// MI455X (gfx1250) — compile-verified
//
#include <hip/hip_runtime.h>
#include <hip/hip_bf16.h>
#include <math.h>

typedef float v2f __attribute__((ext_vector_type(2)));
typedef float v8f __attribute__((ext_vector_type(8)));

#define DK     512      // feature dim (from setup)
#define NBINS  2048     // entropy histogram bins over [0, 3]
#define ENT_HI 3.0f     // entropy <= ln(17) ~ 2.833; ent_bank in [0,1)
#define KMAX   4096     // capacity for per-class selection list
#define BPC    8        // accumulation blocks per class

// ---------------------------------------------------------------- utilities
__global__ void OFTTA_zero_kernel(unsigned* __restrict__ p, size_t n) {
    size_t i = (size_t)blockIdx.x * blockDim.x + threadIdx.x;
    size_t stride = (size_t)gridDim.x * blockDim.x;
    for (; i < n; i += stride) p[i] = 0u;
}

// ------------------------------------------------- WMMA f32 GEMM (shared)
// out[m, c] = sum_d A[m, d] * W[c, d]  (+ bias[c]), A:[M,512], W:[C,512]
// One wave per 16-row M tile; N covered by ceil(C/16) 16-wide tiles.
// Dead columns (>= C) load a clamped-but-valid address and are never stored,
// so they cannot pollute live outputs (WMMA columns are independent).
__global__ __launch_bounds__(128)
void OFTTA_wmma_gemm(const float* __restrict__ A, const float* __restrict__ W,
                     const float* __restrict__ bias, float* __restrict__ out,
                     int M, int C, int hasBias) {
    const int lane = threadIdx.x & 31;
    const int wave = threadIdx.x >> 5;
    const int m0 = blockIdx.x * 64 + wave * 16;
    if (m0 >= M) return;                      // uniform per wave

    const int ln  = lane & 15;                // row (A) / col (B) within tile
    const int kb  = (lane >> 4) << 1;         // K sub-offset: lanes 16-31 -> K+2
    const int rA  = min(m0 + ln, M - 1);
    const float* arow = A + (size_t)rA * DK;
    const int nTiles = (C + 15) >> 4;

    __builtin_prefetch(arow, 0, 0);           // global_prefetch_b8

    for (int t = 0; t < nTiles; ++t) {
        const int n0  = t * 16;
        const int col = min(n0 + ln, C - 1);  // clamp: always valid address
        const float* wrow = W + (size_t)col * DK;
        v8f acc = {};
        #pragma unroll 4
        for (int k = 0; k < DK; k += 4) {
            v2f a, b;
            a.x = arow[k + kb];     a.y = arow[k + kb + 1];
            b.x = wrow[k + kb];     b.y = wrow[k + kb + 1];
            acc = __builtin_amdgcn_wmma_f32_16x16x4_f32(
                false, a, false, b, (short)0, acc, false, false);
        }
        const int ccol = n0 + ln;             // un-clamped store column
        if (ccol < C) {
            const float badd = hasBias ? bias[ccol] : 0.0f;
            const int rowBase = m0 + ((lane >> 4) << 3);
            #pragma unroll
            for (int r = 0; r < 8; ++r) {
                const int row = rowBase + r;
                if (row < M) out[(size_t)row * C + ccol] = acc[r] + badd;
            }
        }
    }
}

// --------------------------------- softmax entropy + argmax, 1 wave / row
__global__ __launch_bounds__(256)
void OFTTA_entropy_kernel(const float* __restrict__ P, float* __restrict__ entNew,
                          int* __restrict__ yhat, int B, int C) {
    const int gtid = blockIdx.x * blockDim.x + threadIdx.x;
    const int row  = gtid >> 5;
    const int lane = gtid & 31;
    if (row >= B) return;

    const float p = (lane < C) ? P[(size_t)row * C + lane] : -INFINITY;

    // max reduce
    float m = p;
    #pragma unroll
    for (int off = 16; off >= 1; off >>= 1)
        m = fmaxf(m, __shfl_xor(m, off, 32));

    const float e  = (lane < C) ? __expf(p - m) : 0.0f;
    float s1 = e;
    float s2 = (lane < C) ? e * p : 0.0f;
    #pragma unroll
    for (int off = 16; off >= 1; off >>= 1) {
        s1 += __shfl_xor(s1, off, 32);
        s2 += __shfl_xor(s2, off, 32);
    }
    const float H = (m + __logf(s1)) - s2 / s1;

    // argmax (tie -> lowest index, matching jnp.argmax)
    float bv = p; int bi = lane;
    #pragma unroll
    for (int off = 16; off >= 1; off >>= 1) {
        float ov = __shfl_xor(bv, off, 32);
        int   oi = __shfl_xor(bi, off, 32);
        if (ov > bv || (ov == bv && oi < bi)) { bv = ov; bi = oi; }
    }
    if (lane == 0) { entNew[row] = H; yhat[row] = bi; }
}

// ------------------------------------------ per-class entropy histograms
__device__ __forceinline__ int ent_bin(float e) {
    int b = (int)(e * ((float)NBINS / ENT_HI));
    return b < 0 ? 0 : (b > NBINS - 1 ? NBINS - 1 : b);
}

__global__ void OFTTA_hist_kernel(const float* __restrict__ entBank,
                                  const int* __restrict__ labels,
                                  const float* __restrict__ entNew,
                                  const int* __restrict__ yhat,
                                  unsigned* __restrict__ hist, int N, int B) {
    int i = blockIdx.x * blockDim.x + threadIdx.x;
    const int stride = gridDim.x * blockDim.x;
    const int total = N + B;
    for (; i < total; i += stride) {
        int c; float e;
        if (i < N) { c = labels[i];    e = entBank[i]; }
        else       { c = yhat[i - N];  e = entNew[i - N]; }
        atomicAdd(&hist[(size_t)c * NBINS + ent_bin(e)], 1u);
    }
}

// --------------------------- per-class threshold bin (Kth smallest entropy)
__global__ void OFTTA_threshold_kernel(const unsigned* __restrict__ hist,
                                       int* __restrict__ Tbin, int* __restrict__ rem,
                                       const int* __restrict__ filterK, int C) {
    const int c = blockIdx.x * blockDim.x + threadIdx.x;
    if (c >= C) return;
    int K = *filterK;
    if (K > KMAX) K = KMAX;
    if (K < 0) K = 0;
    const unsigned* h = hist + (size_t)c * NBINS;
    unsigned below = 0;
    int T = NBINS - 1, r = 0x3fffffff;        // default: include everything
    for (int b = 0; b < NBINS; ++b) {
        const unsigned hb = h[b];
        if (below + hb >= (unsigned)K) { T = b; r = K - (int)below; break; }
        below += hb;
    }
    Tbin[c] = T; rem[c] = r;
}

// -------------------------------------------- build selected-row lists
__global__ void OFTTA_select_kernel(const float* __restrict__ entBank,
                                    const int* __restrict__ labels,
                                    const float* __restrict__ entNew,
                                    const int* __restrict__ yhat,
                                    const int* __restrict__ Tbin,
                                    const int* __restrict__ rem,
                                    int* __restrict__ binTake,
                                    int* __restrict__ selCnt,
                                    int* __restrict__ sel, int N, int B) {
    int i = blockIdx.x * blockDim.x + threadIdx.x;
    const int stride = gridDim.x * blockDim.x;
    const int total = N + B;
    for (; i < total; i += stride) {
        int c; float e;
        if (i < N) { c = labels[i];    e = entBank[i]; }
        else       { c = yhat[i - N];  e = entNew[i - N]; }
        const int b = ent_bin(e);
        const int T = Tbin[c];
        bool take = false;
        if (b < T) take = true;
        else if (b == T) take = (atomicAdd(&binTake[c], 1) < rem[c]);
        if (take) {
            const int s = atomicAdd(&selCnt[c], 1);
            if (s < KMAX) sel[(size_t)c * KMAX + s] = i;
        }
    }
}

// -------- gather selected rows, normalize, sum into Wc (LDS accumulator)
__global__ __launch_bounds__(256)
void OFTTA_accum_kernel(const float* __restrict__ supports,
                        const float* __restrict__ feature,
                        const int* __restrict__ sel, const int* __restrict__ selCnt,
                        float* __restrict__ Wc, int N) {
    __shared__ float acc[DK];
    __shared__ float red[256];
    __shared__ float s_rn;
    const int c   = blockIdx.x / BPC;
    const int bi  = blockIdx.x % BPC;
    const int tid = threadIdx.x;

    acc[tid] = 0.0f; acc[tid + 256] = 0.0f;
    int cnt = selCnt[c];
    if (cnt > KMAX) cnt = KMAX;
    __syncthreads();

    for (int j = bi; j < cnt; j += BPC) {      // uniform across block
        const int row = sel[(size_t)c * KMAX + j];
        const float* src = (row < N) ? (supports + (size_t)row * DK)
                                     : (feature + (size_t)(row - N) * DK);
        const float v0 = src[tid], v1 = src[tid + 256];
        red[tid] = v0 * v0 + v1 * v1;
        __syncthreads();
        #pragma unroll
        for (int off = 128; off > 0; off >>= 1) {
            if (tid < off) red[tid] += red[tid + off];
            __syncthreads();
        }
        if (tid == 0) s_rn = 1.0f / fmaxf(sqrtf(red[0]), 1e-12f);
        __syncthreads();
        const float rn = s_rn;
        acc[tid]       += v0 * rn;
        acc[tid + 256] += v1 * rn;
        __syncthreads();
    }
    atomicAdd(&Wc[(size_t)c * DK + tid],       acc[tid]);
    atomicAdd(&Wc[(size_t)c * DK + tid + 256], acc[tid + 256]);
}

// ------------------------------------- column-normalize class weights
__global__ __launch_bounds__(256)
void OFTTA_normw_kernel(const float* __restrict__ Wc, float* __restrict__ Wn) {
    __shared__ float red[256];
    __shared__ float s_rn;
    const int c = blockIdx.x, tid = threadIdx.x;
    const float v0 = Wc[(size_t)c * DK + tid];
    const float v1 = Wc[(size_t)c * DK + tid + 256];
    red[tid] = v0 * v0 + v1 * v1;
    __syncthreads();
    #pragma unroll
    for (int off = 128; off > 0; off >>= 1) {
        if (tid < off) red[tid] += red[tid + off];
        __syncthreads();
    }
    if (tid == 0) s_rn = 1.0f / fmaxf(sqrtf(red[0]), 1e-12f);
    __syncthreads();
    Wn[(size_t)c * DK + tid]       = v0 * s_rn;
    Wn[(size_t)c * DK + tid + 256] = v1 * s_rn;
}

// ---------------------------------------------------------------- driver
extern "C" void kernel_launch(void* const* d_in, const int* in_sizes, int n_in,
                              void* d_out, int out_size, void* d_ws, size_t ws_size,
                              hipStream_t stream) {
    const float* feature  = (const float*)d_in[0];
    const float* clf_w    = (const float*)d_in[1];
    const float* clf_b    = (const float*)d_in[2];
    const float* supports = (const float*)d_in[3];
    const float* ent_bank = (const float*)d_in[4];
    const int*   labels   = (const int*)d_in[5];
    const int*   filterK  = (const int*)d_in[6];

    const int C = in_sizes[2];            // 17
    const int D = in_sizes[1] / C;        // 512 (kernels assume DK==512)
    const int B = in_sizes[0] / D;        // 4096
    const int N = in_sizes[4];            // 100000

    // workspace layout (all 4-byte types)
    float*    P       = (float*)d_ws;                         // B*C logits
    float*    entNew  = P + (size_t)B * C;                    // B
    int*      yhat    = (int*)(entNew + B);                   // B
    unsigned* hist    = (unsigned*)(yhat + B);                // C*NBINS
    int*      Tbin    = (int*)(hist + (size_t)C * NBINS);     // C
    int*      rem     = Tbin + C;                             // C
    int*      binTake = rem + C;                              // C
    int*      selCnt  = binTake + C;                          // C
    int*      sel     = selCnt + C;                           // C*KMAX
    float*    Wc      = (float*)(sel + (size_t)C * KMAX);     // C*DK
    float*    Wn      = Wc + (size_t)C * DK;                  // C*DK

    // zero hist + counters + sel + Wc (everything accumulated into)
    const size_t zeroN = (size_t)C * NBINS + 4 * (size_t)C +
                         (size_t)C * KMAX + (size_t)C * DK;
    OFTTA_zero_kernel<<<256, 256, 0, stream>>>(hist, zeroN);

    // 1) classifier logits (WMMA f32)
    OFTTA_wmma_gemm<<<(B + 63) / 64, 128, 0, stream>>>(feature, clf_w, clf_b, P, B, C, 1);
    // 2) entropy + pseudo-labels
    OFTTA_entropy_kernel<<<(B * 32 + 255) / 256, 256, 0, stream>>>(P, entNew, yhat, B, C);
    // 3) per-class entropy histograms over the combined bank
    OFTTA_hist_kernel<<<512, 256, 0, stream>>>(ent_bank, labels, entNew, yhat, hist, N, B);
    // 4) per-class K-th-smallest threshold bin
    OFTTA_threshold_kernel<<<1, 32, 0, stream>>>(hist, Tbin, rem, filterK, C);
    // 5) build selected-row lists (cap ties in threshold bin)
    OFTTA_select_kernel<<<512, 256, 0, stream>>>(ent_bank, labels, entNew, yhat,
                                                 Tbin, rem, binTake, selCnt, sel, N, B);
    // 6) gather + normalize + per-class sum
    OFTTA_accum_kernel<<<C * BPC, 256, 0, stream>>>(supports, feature, sel, selCnt, Wc, N);
    // 7) normalize class weight columns
    OFTTA_normw_kernel<<<C, 256, 0, stream>>>(Wc, Wn);
    // 8) final projection (WMMA f32) -> d_out
    OFTTA_wmma_gemm<<<(B + 63) / 64, 128, 0, stream>>>(feature, Wn, nullptr,
                                                       (float*)d_out, B, C, 0);
}